// SimpleAdapterModel_29360396436109
// MI455X (gfx1250) — compile-verified
//
#include <hip/hip_runtime.h>
#include <hip/hip_bf16.h>

// Problem constants (from reference)
#define BATCH        512
#define NUM_PROPS    128
#define HDIM         512
#define ODIM         2
#define TOTAL_PAIRS  (BATCH * NUM_PROPS)   // 65536

// Mapping: 1 wave32 per (b,p) pair, PAIRS_PER_WAVE pairs per wave,
// 8 waves (256 threads) per block.
#define WAVES_PER_BLOCK 8
#define PAIRS_PER_WAVE  4
#define PAIRS_PER_BLOCK (WAVES_PER_BLOCK * PAIRS_PER_WAVE)   // 32
#define SLICE_FLOATS    HDIM                                  // 512 f32 = 2KB per buffer

// 4 async b128 loads move one 2KB hidden vector into LDS.
// Per lane: 16B at (lane*16 + j*512); the instruction offset applies to BOTH
// the global and the LDS address (CDNA5 ISA).
// th:TH_LOAD_NT — hidden states are streamed exactly once (134 MB); keep them
// from evicting the L2-resident 41 MB weights table (192 MB L2).
#define ISSUE_ASYNC_HIDDEN(ldsaddr, gaddr)                                               \
  do {                                                                                   \
    asm volatile("global_load_async_to_lds_b128 %0, %1, off th:TH_LOAD_NT"               \
                 :: "v"(ldsaddr), "v"(gaddr) : "memory");                                \
    asm volatile("global_load_async_to_lds_b128 %0, %1, off offset:512 th:TH_LOAD_NT"    \
                 :: "v"(ldsaddr), "v"(gaddr) : "memory");                                \
    asm volatile("global_load_async_to_lds_b128 %0, %1, off offset:1024 th:TH_LOAD_NT"   \
                 :: "v"(ldsaddr), "v"(gaddr) : "memory");                                \
    asm volatile("global_load_async_to_lds_b128 %0, %1, off offset:1536 th:TH_LOAD_NT"   \
                 :: "v"(ldsaddr), "v"(gaddr) : "memory");                                \
  } while (0)

__global__ __launch_bounds__(256)
void adapter_gather_gemv_kernel(const float* __restrict__ hidden,   // (B,P,H)
                                const int*   __restrict__ props,    // (B,P)
                                const float* __restrict__ mask,     // (B,P)
                                const float* __restrict__ weights,  // (NPROP,H,2)
                                const float* __restrict__ bias,     // (NPROP,2)
                                float* __restrict__ out)            // (B,P,2)
{
  // Double-buffered hidden-vector staging: 8 waves * 2 * 2KB = 32KB LDS.
  __shared__ float smem[WAVES_PER_BLOCK * 2 * SLICE_FLOATS];

  const int lane = threadIdx.x & 31;
  const int wv   = __builtin_amdgcn_readfirstlane((int)(threadIdx.x >> 5)); // wave-uniform
  const int pair0 = blockIdx.x * PAIRS_PER_BLOCK + wv * PAIRS_PER_WAVE;

  // Per-lane LDS byte offsets (generic-pointer low 32 bits == wave-relative LDS addr)
  float* bufA = &smem[(wv * 2 + 0) * SLICE_FLOATS];
  float* bufB = &smem[(wv * 2 + 1) * SLICE_FLOATS];
  const unsigned ldsA = (unsigned)(unsigned long long)(bufA + lane * 4); // lane*16B
  const unsigned ldsB = (unsigned)(unsigned long long)(bufB + lane * 4);

  // Prologue: start async fill of buffer 0 with pair0's hidden vector.
  {
    unsigned long long g0 =
        (unsigned long long)(hidden + (size_t)pair0 * HDIM + lane * 4);
    ISSUE_ASYNC_HIDDEN(ldsA, g0);
  }

#pragma unroll
  for (int t = 0; t < PAIRS_PER_WAVE; ++t) {
    const int pair = pair0 + t;

    // Uniform property index -> scalar load; weight base lives in SGPRs (GVS gather).
    const int idx = props[pair];
    const float* __restrict__ Wb = weights + (size_t)idx * (HDIM * ODIM);

    if (t + 1 < PAIRS_PER_WAVE) {
      // Buffer (t+1)&1 was last ds-read at iteration t-1; async LDS writes are
      // unordered vs DS ops, so drain DScnt before overwriting it.
      asm volatile("s_wait_dscnt 0" ::: "memory");
      unsigned long long gn =
          (unsigned long long)(hidden + (size_t)(pair + 1) * HDIM + lane * 4);
      const unsigned ldsn = ((t + 1) & 1) ? ldsB : ldsA;
      ISSUE_ASYNC_HIDDEN(ldsn, gn);
      // 8 async ops outstanding; async loads complete in order, so <=4 means
      // the 4 belonging to the CURRENT pair are done.
      asm volatile("s_wait_asynccnt 4" ::: "memory");
    } else {
      asm volatile("s_wait_asynccnt 0" ::: "memory");
    }

    const float* __restrict__ hb = ((t & 1) ? bufB : bufA);

    // Lane owns h rows {4*lane + 128*j + k : j=0..3, k=0..3} (matches async layout).
    float acc0 = 0.0f, acc1 = 0.0f;
#pragma unroll
    for (int j = 0; j < 4; ++j) {
      const float4 h4 = *(const float4*)(hb + lane * 4 + j * 128);            // ds_load_b128
      const float4* wp = (const float4*)(Wb + 8 * lane + 256 * j);            // coalesced b128 gather (RT: stay in L2)
      const float4 wlo = wp[0];  // rows r0,r1: (W[r0,0],W[r0,1],W[r1,0],W[r1,1])
      const float4 whi = wp[1];  // rows r2,r3
      acc0 += h4.x * wlo.x + h4.y * wlo.z + h4.z * whi.x + h4.w * whi.z;
      acc1 += h4.x * wlo.y + h4.y * wlo.w + h4.z * whi.y + h4.w * whi.w;
    }

    // wave32 butterfly reduction (warpSize == 32 on gfx1250)
#pragma unroll
    for (int off = 16; off > 0; off >>= 1) {
      acc0 += __shfl_xor(acc0, off, 32);
      acc1 += __shfl_xor(acc1, off, 32);
    }

    if (lane == 0) {
      const float m = mask[pair];
      // Output written once, never re-read: non-temporal stores.
      __builtin_nontemporal_store((acc0 + bias[2 * idx + 0]) * m, &out[2 * pair + 0]);
      __builtin_nontemporal_store((acc1 + bias[2 * idx + 1]) * m, &out[2 * pair + 1]);
    }
  }
}

extern "C" void kernel_launch(void* const* d_in, const int* in_sizes, int n_in,
                              void* d_out, int out_size, void* d_ws, size_t ws_size,
                              hipStream_t stream) {
  const float* hidden  = (const float*)d_in[0];  // (512,128,512) f32
  const int*   props   = (const int*)  d_in[1];  // (512,128) int
  const float* mask    = (const float*)d_in[2];  // (512,128) f32
  const float* weights = (const float*)d_in[3];  // (10000,512,2) f32
  const float* bias    = (const float*)d_in[4];  // (10000,2) f32
  float* out = (float*)d_out;                    // (512,128,2) f32

  dim3 grid(TOTAL_PAIRS / PAIRS_PER_BLOCK);      // 2048 blocks
  dim3 block(WAVES_PER_BLOCK * 32);              // 256 threads = 8 wave32
  adapter_gather_gemv_kernel<<<grid, block, 0, stream>>>(hidden, props, mask,
                                                         weights, bias, out);
}